// PatchAugmenter_32315333935135
// MI455X (gfx1250) — compile-verified
//
#include <hip/hip_runtime.h>
#include <hip/hip_bf16.h>

typedef __attribute__((ext_vector_type(2))) float v2f;
typedef __attribute__((ext_vector_type(8))) float v8f;

// ---------------------------------------------------------------------------
// Kernel 1: rotate (bilinear or nearest) + flip for (C,256,256) fp32 images.
// One thread per OUTPUT pixel -> coalesced stores; gathers are L2-resident
// (each 256KB source channel << 192MB L2). Flips are folded into which output
// coordinate we evaluate the affine grid at (flip-after-rotate == evaluating
// the rotated image at the flipped index).
// ---------------------------------------------------------------------------
__global__ void rot_flip_img_kernel(const float* __restrict__ src,
                                    float* __restrict__ dst,
                                    const float* __restrict__ angles,
                                    const int* __restrict__ flip_h,
                                    const int* __restrict__ flip_v,
                                    int C, int bilinear, int total)
{
    int id = blockIdx.x * blockDim.x + threadIdx.x;
    if (id >= total) return;

    const int W = 256, H = 256;
    int x  = id & 255;
    int y  = (id >> 8) & 255;
    int c  = (id >> 16) % C;
    int bk = id / (65536 * C);

    int fh = flip_h[bk] != 0;
    int fv = flip_v[bk] != 0;
    int xs = fh ? (W - 1 - x) : x;
    int ys = fv ? (H - 1 - y) : y;

    float ang = angles[bk];
    float cs = cosf(ang), sn = sinf(ang);
    float gx = (2.0f * (float)xs + 1.0f) * (1.0f / (float)W) - 1.0f;
    float gy = (2.0f * (float)ys + 1.0f) * (1.0f / (float)H) - 1.0f;
    float xin = cs * gx - sn * gy;
    float yin = sn * gx + cs * gy;
    float xp = (xin + 1.0f) * ((float)W * 0.5f) - 0.5f;
    float yp = (yin + 1.0f) * ((float)H * 0.5f) - 0.5f;

    const float* img = src + ((size_t)bk * C + c) * (size_t)(H * W);
    float v;

    if (bilinear) {
        float x0f = floorf(xp), y0f = floorf(yp);
        float wx = xp - x0f, wy = yp - y0f;
        int x0 = (int)x0f, y0 = (int)y0f;
        int x1 = x0 + 1,   y1 = y0 + 1;

        auto tap = [&](int yi, int xi) -> float {
            bool ok = (yi >= 0) & (yi < H) & (xi >= 0) & (xi < W);
            int yc = min(max(yi, 0), H - 1);
            int xc = min(max(xi, 0), W - 1);
            float t = img[(size_t)yc * W + xc];
            return ok ? t : 0.0f;
        };
        float v00 = tap(y0, x0), v01 = tap(y0, x1);
        float v10 = tap(y1, x0), v11 = tap(y1, x1);
        v = (v00 * (1.0f - wx) + v01 * wx) * (1.0f - wy)
          + (v10 * (1.0f - wx) + v11 * wx) * wy;
    } else {
        // jnp.round == round-half-even == rintf (RNE default rounding)
        int xi = (int)rintf(xp);
        int yi = (int)rintf(yp);
        bool ok = (yi >= 0) & (yi < H) & (xi >= 0) & (xi < W);
        int yc = min(max(yi, 0), H - 1);
        int xc = min(max(xi, 0), W - 1);
        float t = img[(size_t)yc * W + xc];
        v = ok ? t : 0.0f;
    }
    dst[id] = v;
}

// ---------------------------------------------------------------------------
// Kernel 2: feature rotation+flip as a WMMA matmul.
//   out[t, d] = sum_s S[t, s] * F[s, d],  per (b,k);  T=256 (16x16 grid), D=768
// S = bilinear interpolation matrix (4 nonzeros/row), flips folded into the
// grid coordinate used for output row t. S is never materialized: each lane
// owns tile-row m = lane&15, precomputes its 4 source columns + weights, and
// synthesizes its A-fragment (V_WMMA_F32_16X16X4_F32 layout: lanes 0-15 hold
// K=k0..k0+1, lanes 16-31 K=k0+2..k0+3) branchlessly every k-step.
// One wave per (image, mtile, group-of-4 ntiles): A reused across 4 WMMAs.
// ---------------------------------------------------------------------------
__global__ void feat_rot_wmma_kernel(const float* __restrict__ feat,
                                     float* __restrict__ out,
                                     const float* __restrict__ angles,
                                     const int* __restrict__ flip_h,
                                     const int* __restrict__ flip_v)
{
    const int T = 256, D = 768, HH = 16;

    int job   = blockIdx.x;        // nimg * 16 mtiles * 12 ngroups
    int ng    = job % 12;
    int mtile = (job / 12) & 15;
    int bk    = job / (12 * 16);

    int lane  = threadIdx.x;       // 0..31, wave32
    int m     = lane & 15;         // A row / B column within tile
    int khalf = lane >> 4;         // K sub-slot per fp32 WMMA layout

    // output spatial position for this lane's A row
    int s = mtile * 16 + m;
    int y = s >> 4, x = s & 15;
    int fh = flip_h[bk] != 0;
    int fv = flip_v[bk] != 0;
    int ysrc = fv ? (HH - 1 - y) : y;
    int xsrc = fh ? (HH - 1 - x) : x;

    float ang = angles[bk];
    float cs = cosf(ang), sn = sinf(ang);
    float gx = (2.0f * (float)xsrc + 1.0f) * (1.0f / 16.0f) - 1.0f;
    float gy = (2.0f * (float)ysrc + 1.0f) * (1.0f / 16.0f) - 1.0f;
    float xin = cs * gx - sn * gy;
    float yin = sn * gx + cs * gy;
    float xp = (xin + 1.0f) * 8.0f - 0.5f;
    float yp = (yin + 1.0f) * 8.0f - 0.5f;

    float x0f = floorf(xp), y0f = floorf(yp);
    float wx = xp - x0f, wy = yp - y0f;
    int x0 = (int)x0f, y0 = (int)y0f;
    int x1 = x0 + 1,   y1 = y0 + 1;

    int   c0, c1, c2, c3;
    float w0, w1, w2, w3;
    auto corner = [&](int yi, int xi, float w, int& col, float& wt) {
        bool ok = (yi >= 0) & (yi < HH) & (xi >= 0) & (xi < HH);
        int yc = min(max(yi, 0), HH - 1);
        int xc = min(max(xi, 0), HH - 1);
        col = yc * HH + xc;
        wt  = ok ? w : 0.0f;
    };
    corner(y0, x0, (1.0f - wx) * (1.0f - wy), c0, w0);
    corner(y0, x1, wx * (1.0f - wy),          c1, w1);
    corner(y1, x0, (1.0f - wx) * wy,          c2, w2);
    corner(y1, x1, wx * wy,                   c3, w3);

    const float* F = feat + (size_t)bk * T * D;
    int n     = lane & 15;
    int nbase = ng * 64;

    v8f acc0 = {}; v8f acc1 = {}; v8f acc2 = {}; v8f acc3 = {};

    for (int kb = 0; kb < 64; ++kb) {
        int kA  = kb * 4 + khalf * 2;
        int kA1 = kA + 1;

        v2f a;
        a.x = (c0 == kA  ? w0 : 0.0f) + (c1 == kA  ? w1 : 0.0f)
            + (c2 == kA  ? w2 : 0.0f) + (c3 == kA  ? w3 : 0.0f);
        a.y = (c0 == kA1 ? w0 : 0.0f) + (c1 == kA1 ? w1 : 0.0f)
            + (c2 == kA1 ? w2 : 0.0f) + (c3 == kA1 ? w3 : 0.0f);

        const float* r0 = F + (size_t)kA  * D + nbase + n;
        const float* r1 = F + (size_t)kA1 * D + nbase + n;
        v2f b0; b0.x = r0[0];  b0.y = r1[0];
        v2f b1; b1.x = r0[16]; b1.y = r1[16];
        v2f b2; b2.x = r0[32]; b2.y = r1[32];
        v2f b3; b3.x = r0[48]; b3.y = r1[48];

        acc0 = __builtin_amdgcn_wmma_f32_16x16x4_f32(false, a, false, b0, (short)0, acc0, false, false);
        acc1 = __builtin_amdgcn_wmma_f32_16x16x4_f32(false, a, false, b1, (short)0, acc1, false, false);
        acc2 = __builtin_amdgcn_wmma_f32_16x16x4_f32(false, a, false, b2, (short)0, acc2, false, false);
        acc3 = __builtin_amdgcn_wmma_f32_16x16x4_f32(false, a, false, b3, (short)0, acc3, false, false);
    }

    // C/D layout: VGPR r -> rows {r, r+8} split across lane halves, N = lane&15
    float* O = out + (size_t)bk * T * D;
    for (int r = 0; r < 8; ++r) {
        int row = mtile * 16 + r + khalf * 8;
        size_t base = (size_t)row * D + nbase + n;
        O[base +  0] = acc0[r];
        O[base + 16] = acc1[r];
        O[base + 32] = acc2[r];
        O[base + 48] = acc3[r];
    }
}

extern "C" void kernel_launch(void* const* d_in, const int* in_sizes, int n_in,
                              void* d_out, int out_size, void* d_ws, size_t ws_size,
                              hipStream_t stream) {
    (void)in_sizes; (void)n_in; (void)out_size; (void)d_ws; (void)ws_size;

    const float* patches = (const float*)d_in[0];   // (4,16,3,256,256) f32
    const float* labels  = (const float*)d_in[1];   // (4,16,1,256,256) f32
    const float* feats   = (const float*)d_in[2];   // (4,16,256,768)   f32
    const float* angles  = (const float*)d_in[3];   // (4,16)           f32
    const int*   fh      = (const int*)d_in[4];     // (4,16) bool->int
    const int*   fv      = (const int*)d_in[5];     // (4,16) bool->int

    const int B = 4, K = 16, C = 3, H = 256, W = 256, T = 256, D = 768;
    float* out = (float*)d_out;
    size_t apN = (size_t)B * K * C * H * W;   // 12,582,912
    size_t alN = (size_t)B * K * 1 * H * W;   //  4,194,304
    float* ap = out;
    float* al = out + apN;
    float* fs = out + apN + alN;

    int total_p = (int)apN;
    rot_flip_img_kernel<<<(total_p + 255) / 256, 256, 0, stream>>>(
        patches, ap, angles, fh, fv, C, /*bilinear=*/1, total_p);

    int total_l = (int)alN;
    rot_flip_img_kernel<<<(total_l + 255) / 256, 256, 0, stream>>>(
        labels, al, angles, fh, fv, 1, /*bilinear=*/0, total_l);

    int jobs = B * K * 16 * 12;               // 12,288 single-wave jobs
    feat_rot_wmma_kernel<<<jobs, 32, 0, stream>>>(feats, fs, angles, fh, fv);
}